// GlobalLocalAttention_7000796692537
// MI455X (gfx1250) — compile-verified
//
#include <hip/hip_runtime.h>

// ---------------------------------------------------------------------------
// Swin window attention, fp32, CDNA5 (gfx1250) wave32 + V_WMMA_F32_16X16X4_F32
// ---------------------------------------------------------------------------

typedef float v2f __attribute__((ext_vector_type(2)));
typedef float v8f __attribute__((ext_vector_type(8)));

#define NHh   8
#define HDd   64
#define DIMm  512
#define Bb    16
#define NTOK  1024
#define HEADSZ (Bb * NHh * NTOK * HDd)   // 8,388,608 floats per q/k/v buffer

__device__ __forceinline__ v8f wmma4(v2f a, v2f b, v8f c) {
    // D = A(16x4 f32) x B(4x16 f32) + C(16x16 f32)
    return __builtin_amdgcn_wmma_f32_16x16x4_f32(
        /*neg_a=*/false, a, /*neg_b=*/false, b,
        /*c_mod=*/(short)0, c, /*reuse_a=*/false, /*reuse_b=*/false);
}

// ---------------------------------------------------------------------------
// Kernel 1: qkv = x @ qkv_w^T + qkv_b  -> head-major q/k/v (q pre-scaled)
// One 16(token) x 16(channel) tile per wave. M=16384, N=1536, K=512.
// ---------------------------------------------------------------------------
__global__ void qkv_gemm_kernel(const float* __restrict__ x,
                                const float* __restrict__ w,
                                const float* __restrict__ bias,
                                float* __restrict__ qkv) {
    const int lane = threadIdx.x & 31;
    const int lx   = lane & 15;
    const int half = lane >> 4;

    int gw = blockIdx.x * (blockDim.x >> 5) + (threadIdx.x >> 5);
    const int ct = gw % 96;  gw /= 96;   // 1536/16 output-channel tiles
    const int mt = gw % 64;  gw /= 64;   // 1024/16 token tiles
    const int b  = gw;                   // batch

    // A frag: a[v] = x[b, mt*16+lx, k + 2*half + v]
    const float* arow = x + ((size_t)(b * NTOK + mt * 16 + lx)) * DIMm + 2 * half;
    // B frag: b[v] = qkv_w[ct*16+lx, k + 2*half + v]   (out = x @ W^T)
    const float* brow = w + ((size_t)(ct * 16 + lx)) * DIMm + 2 * half;

    v8f acc = {};
#pragma unroll 8
    for (int k = 0; k < DIMm; k += 4) {
        v2f a  = *(const v2f*)(arow + k);
        v2f bf = *(const v2f*)(brow + k);
        acc = wmma4(a, bf, acc);
    }

    // epilogue: scatter into q/k/v  [B][NH][N][HD], scale q by HD^-0.5
    const int   c     = ct * 16 + lx;          // column per lane (all 8 rows)
    const float bv    = bias[c];
    const int   which = c >> 9;                // 0=q 1=k 2=v
    const int   h     = (c >> 6) & 7;
    const int   d     = c & 63;
    const float scale = (which == 0) ? 0.125f : 1.0f;

    float* dst = qkv + (size_t)which * HEADSZ
               + ((size_t)(b * NHh + h) * NTOK) * HDd + d;
#pragma unroll
    for (int r = 0; r < 8; ++r) {
        const int row = mt * 16 + r + 8 * half;
        dst[(size_t)row * HDd] = (acc[r] + bv) * scale;
    }
}

// ---------------------------------------------------------------------------
// Kernel 2: fused attention (flash-style, scores never hit memory).
// One wave handles 16 query rows of one (b, h). 2048 WMMAs per wave.
// ---------------------------------------------------------------------------
__global__ void attn_kernel(const float* __restrict__ qkv,
                            const float* __restrict__ bias_table,
                            const int*   __restrict__ rel_index,
                            float*       __restrict__ ctx) {
    __shared__ float pbuf[8][16 * 16];       // per-wave P-tile transpose stage
    const int lane = threadIdx.x & 31;
    const int lx   = lane & 15;
    const int half = lane >> 4;
    const int wv   = threadIdx.x >> 5;
    float* pb = pbuf[wv];

    int gw = blockIdx.x * (blockDim.x >> 5) + wv;
    const int qt = gw % 64;  gw /= 64;       // query tile
    const int h  = gw % 8;   gw /= 8;        // head
    const int b  = gw;                       // batch

    const size_t headoff = ((size_t)(b * NHh + h) * NTOK) * HDd;
    const float* qp = qkv + headoff;                     // q (pre-scaled)
    const float* kp = qkv + (size_t)HEADSZ + headoff;    // k
    const float* vp = qkv + 2 * (size_t)HEADSZ + headoff;// v

    // Q tile (16x64) as 16 A-fragments, kept in registers
    v2f qf[16];
    {
        const float* qrow = qp + (size_t)(qt * 16 + lx) * HDd + 2 * half;
#pragma unroll
        for (int kk = 0; kk < 16; ++kk) qf[kk] = *(const v2f*)(qrow + kk * 4);
    }

    v8f cacc[4];
#pragma unroll
    for (int t = 0; t < 4; ++t) cacc[t] = (v8f){};
    float mrow[8], lrow[8];
#pragma unroll
    for (int r = 0; r < 8; ++r) { mrow[r] = -3.0e38f; lrow[r] = 0.0f; }

    for (int j = 0; j < 64; ++j) {           // key tiles of 16
        // ----- S = Q . K^T  (16x16 tile, K-dim 64) -----
        v8f s = (v8f){};
        const float* krow = kp + (size_t)(j * 16 + lx) * HDd + 2 * half;
#pragma unroll
        for (int kk = 0; kk < 16; ++kk) {
            v2f bf = *(const v2f*)(krow + kk * 4);
            s = wmma4(qf[kk], bf, s);
        }

        // ----- relative position bias gather -----
        const int kcol = j * 16 + lx;
#pragma unroll
        for (int r = 0; r < 8; ++r) {
            const int qrow_ = qt * 16 + r + 8 * half;
            const int idx   = rel_index[qrow_ * NTOK + kcol];
            s[r] += bias_table[idx * NHh + h];
        }

        // ----- online softmax (rows live in 16-lane halves) -----
        float p[8], alpha[8];
#pragma unroll
        for (int r = 0; r < 8; ++r) {
            float tmax = s[r];
            tmax = fmaxf(tmax, __shfl_xor(tmax, 1, 32));
            tmax = fmaxf(tmax, __shfl_xor(tmax, 2, 32));
            tmax = fmaxf(tmax, __shfl_xor(tmax, 4, 32));
            tmax = fmaxf(tmax, __shfl_xor(tmax, 8, 32));
            const float mnew = fmaxf(mrow[r], tmax);
            alpha[r] = __expf(mrow[r] - mnew);
            const float pv = __expf(s[r] - mnew);
            float rs = pv;
            rs += __shfl_xor(rs, 1, 32);
            rs += __shfl_xor(rs, 2, 32);
            rs += __shfl_xor(rs, 4, 32);
            rs += __shfl_xor(rs, 8, 32);
            lrow[r] = lrow[r] * alpha[r] + rs;
            mrow[r] = mnew;
            p[r] = pv;
        }
#pragma unroll
        for (int t = 0; t < 4; ++t)
#pragma unroll
            for (int r = 0; r < 8; ++r) cacc[t][r] *= alpha[r];

        // ----- transpose P (C-layout -> A-layout) through per-wave LDS -----
#pragma unroll
        for (int r = 0; r < 8; ++r) pb[(r + 8 * half) * 16 + lx] = p[r];
        v2f af[4];
#pragma unroll
        for (int kk = 0; kk < 4; ++kk)
            af[kk] = *(const v2f*)(pb + lx * 16 + kk * 4 + 2 * half);

        // ----- ctx += P . V  (K-dim = 16 keys, N-dim = 64 head dims) -----
#pragma unroll
        for (int t = 0; t < 4; ++t) {
            const float* vcol = vp + (size_t)(j * 16) * HDd + t * 16 + lx;
#pragma unroll
            for (int kk = 0; kk < 4; ++kk) {
                const int kr = kk * 4 + 2 * half;
                v2f bf;
                bf[0] = vcol[(size_t)kr * HDd];
                bf[1] = vcol[(size_t)(kr + 1) * HDd];
                cacc[t] = wmma4(af[kk], bf, cacc[t]);
            }
        }
    }

    // ----- normalize and store ctx [B][N][NH*HD] -----
#pragma unroll
    for (int r = 0; r < 8; ++r) lrow[r] = 1.0f / lrow[r];
#pragma unroll
    for (int t = 0; t < 4; ++t) {
#pragma unroll
        for (int r = 0; r < 8; ++r) {
            const int n = qt * 16 + r + 8 * half;
            ctx[((size_t)(b * NTOK + n)) * DIMm + h * HDd + t * 16 + lx] =
                cacc[t][r] * lrow[r];
        }
    }
}

// ---------------------------------------------------------------------------
// Kernel 3: out = ctx @ proj_w^T + proj_b, stored transposed as [B, DIM, N].
// M-dim = output channels so stores along N are lane-contiguous.
// ---------------------------------------------------------------------------
__global__ void proj_gemm_kernel(const float* __restrict__ ctx,
                                 const float* __restrict__ w,
                                 const float* __restrict__ bias,
                                 float* __restrict__ out) {
    const int lane = threadIdx.x & 31;
    const int lx   = lane & 15;
    const int half = lane >> 4;

    int gw = blockIdx.x * (blockDim.x >> 5) + (threadIdx.x >> 5);
    const int nt = gw % 64;  gw /= 64;   // token tiles
    const int ct = gw % 32;  gw /= 32;   // 512/16 channel tiles
    const int b  = gw;

    // A frag: proj_w[ct*16+lx, k...]; B frag: ctx[b, nt*16+lx, k...]
    const float* arow = w   + (size_t)(ct * 16 + lx) * DIMm + 2 * half;
    const float* brow = ctx + (size_t)(b * NTOK + nt * 16 + lx) * DIMm + 2 * half;

    v8f acc = {};
#pragma unroll 8
    for (int k = 0; k < DIMm; k += 4) {
        v2f a  = *(const v2f*)(arow + k);
        v2f bf = *(const v2f*)(brow + k);
        acc = wmma4(a, bf, acc);
    }

    const int n = nt * 16 + lx;
#pragma unroll
    for (int r = 0; r < 8; ++r) {
        const int c = ct * 16 + r + 8 * half;
        out[((size_t)(b * DIMm + c)) * NTOK + n] = acc[r] + bias[c];
    }
}

// ---------------------------------------------------------------------------
extern "C" void kernel_launch(void* const* d_in, const int* in_sizes, int n_in,
                              void* d_out, int out_size, void* d_ws, size_t ws_size,
                              hipStream_t stream) {
    const float* x          = (const float*)d_in[0];
    const float* qkv_w      = (const float*)d_in[1];
    const float* qkv_b      = (const float*)d_in[2];
    const float* proj_w     = (const float*)d_in[3];
    const float* proj_b     = (const float*)d_in[4];
    const float* bias_table = (const float*)d_in[5];
    const int*   rel_index  = (const int*)d_in[6];
    float*       out        = (float*)d_out;

    float* wsf = (float*)d_ws;
    float* qkv = wsf;                         // q | k | v, 3*HEADSZ floats
    float* ctx = wsf + 3 * (size_t)HEADSZ;    // [B][N][512]

    dim3 blk(256);  // 8 waves per block

    // 16*64*96 = 98304 tiles / 8 waves
    qkv_gemm_kernel<<<12288, blk, 0, stream>>>(x, qkv_w, qkv_b, qkv);
    // 16*8*64 = 8192 waves / 8
    attn_kernel<<<1024, blk, 0, stream>>>(qkv, bias_table, rel_index, ctx);
    // 16*32*64 = 32768 tiles / 8
    proj_gemm_kernel<<<4096, blk, 0, stream>>>(ctx, proj_w, proj_b, out);
}